// RGCNEncoder_77627238908624
// MI455X (gfx1250) — compile-verified
//
#include <hip/hip_runtime.h>

#define H 128
#define HT 8  // H/16 column tiles

typedef float v2f __attribute__((ext_vector_type(2)));
typedef float v8f __attribute__((ext_vector_type(8)));

// ---------------------------------------------------------------------------
// w0[i][o] = sum_b comp[b] * basis[b][i][o]      (one 128x128 matrix)
// ---------------------------------------------------------------------------
__global__ void w_combine_kernel(const float* __restrict__ basis,
                                 const float* __restrict__ comp,
                                 float* __restrict__ w0) {
    int idx = blockIdx.x * blockDim.x + threadIdx.x;   // 0..H*H-1
    float acc = 0.0f;
#pragma unroll
    for (int b = 0; b < 30; ++b)
        acc += comp[b] * basis[(size_t)b * H * H + idx];
    w0[idx] = acc;
}

// ---------------------------------------------------------------------------
// degree count + reciprocal (scatter-mean weights)
// ---------------------------------------------------------------------------
__global__ void deg_count_kernel(const int* __restrict__ dst,
                                 float* __restrict__ deg, int E) {
    int e = blockIdx.x * blockDim.x + threadIdx.x;
    if (e < E) atomicAdd(&deg[dst[e]], 1.0f);
}

__global__ void deg_recip_kernel(float* __restrict__ deg, int n) {
    int i = blockIdx.x * blockDim.x + threadIdx.x;
    if (i < n) deg[i] = 1.0f / fmaxf(deg[i], 1.0f);
}

// ---------------------------------------------------------------------------
// C[M x 128] = A[M x 128] * W[128 x 128]  in fp32 via v_wmma_f32_16x16x4_f32
// One wave per 16-row tile; 8 col tiles (v8f accumulators) per wave.
// W staged in LDS packed as float2 = (W[2k][o], W[2k+1][o]) so every B
// fragment is one ds_load_b64 and every A fragment one 8-byte global load.
// ---------------------------------------------------------------------------
__global__ void __launch_bounds__(256)
gemm_wmma_kernel(const float* __restrict__ A, const float* __restrict__ W,
                 float* __restrict__ C, int nTiles) {
    __shared__ v2f lds2[(H / 2) * H];   // 64 KB
    const int tid = threadIdx.x;

    // stage + K-pair-pack W into LDS
    for (int i = tid; i < (H / 2) * H; i += 256) {
        int kp = i >> 7;        // K pair index 0..63
        int o  = i & (H - 1);   // output column
        v2f v;
        v.x = W[(2 * kp) * H + o];
        v.y = W[(2 * kp + 1) * H + o];
        lds2[i] = v;
    }
    __syncthreads();

    const int wave = tid >> 5;
    const int lane = tid & 31;
    const int tile = blockIdx.x * 8 + wave;
    if (tile >= nTiles) return;                 // uniform per wave: EXEC stays all-ones

    const int lanelo = lane & 15;
    const int koff   = (lane >> 4) << 1;        // 0 (lanes 0-15) or 2 (lanes 16-31)
    const float* arow = A + (size_t)(tile * 16 + lanelo) * H;

    v8f acc[HT] = {};

    for (int k0 = 0; k0 < H; k0 += 4) {
        // A fragment: lane<16 -> (K=k0,k0+1); lane>=16 -> (K=k0+2,k0+3) of row M=lanelo
        v2f a = *(const v2f*)(arow + k0 + koff);
        const v2f* brow = &lds2[((k0 + koff) >> 1) * H + lanelo];
#pragma unroll
        for (int ct = 0; ct < HT; ++ct) {
            v2f b = brow[ct * 16];   // (W[k0+koff][o], W[k0+koff+1][o]), o = ct*16+lanelo
            acc[ct] = __builtin_amdgcn_wmma_f32_16x16x4_f32(
                false, a, false, b, (short)0, acc[ct], false, false);
        }
    }

    // C/D layout: VGPR r -> row M = r (lanes 0-15) or r+8 (lanes 16-31), col N = lane&15
    const int rowBase = tile * 16 + ((lane >> 4) << 3);
#pragma unroll
    for (int r = 0; r < 8; ++r) {
        float* crow = C + (size_t)(rowBase + r) * H + lanelo;
#pragma unroll
        for (int ct = 0; ct < HT; ++ct) crow[ct * 16] = acc[ct][r];
    }
}

// ---------------------------------------------------------------------------
// out[dst[e]] += y[src[e]] * invdeg[dst[e]]   (32 lanes per edge, float4 each)
// ---------------------------------------------------------------------------
__global__ void scatter_kernel(const int* __restrict__ src,
                               const int* __restrict__ dst,
                               const float* __restrict__ y,
                               const float* __restrict__ invdeg,
                               float* __restrict__ out, int E) {
    long gid = (long)blockIdx.x * blockDim.x + threadIdx.x;
    int e    = (int)(gid >> 5);
    int lane = (int)(gid & 31);
    if (e >= E) return;
    int s = src[e];
    int d = dst[e];
    float w = invdeg[d];
    float4 v = *(const float4*)(y + (size_t)s * H + lane * 4);
    float* o = out + (size_t)d * H + lane * 4;
    atomicAdd(o + 0, v.x * w);
    atomicAdd(o + 1, v.y * w);
    atomicAdd(o + 2, v.z * w);
    atomicAdd(o + 3, v.w * w);
}

// ---------------------------------------------------------------------------
// x = relu(x + bias[col])
// ---------------------------------------------------------------------------
__global__ void finalize_kernel(float* __restrict__ x,
                                const float* __restrict__ bias, long n) {
    long i = (long)blockIdx.x * blockDim.x + threadIdx.x;
    if (i < n) {
        int c = (int)(i & (H - 1));
        x[i] = fmaxf(x[i] + bias[c], 0.0f);
    }
}

// ---------------------------------------------------------------------------
extern "C" void kernel_launch(void* const* d_in, const int* in_sizes, int n_in,
                              void* d_out_v, int out_size, void* d_ws, size_t ws_size,
                              hipStream_t stream) {
    (void)n_in; (void)out_size; (void)ws_size;

    const float* emb_drug = (const float*)d_in[0];   // [Nd, H]
    const float* emb_dis  = (const float*)d_in[1];   // [Ns, H]
    const float* basis    = (const float*)d_in[2];   // [2, 30, H, H]
    const float* comp     = (const float*)d_in[3];   // [2, 8, 30]
    const float* root     = (const float*)d_in[4];   // [2, H, H]
    const float* bias     = (const float*)d_in[5];   // [2, H]
    const int*   edd      = (const int*)d_in[6];     // [2, E] drug->disease
    const int*   edsd     = (const int*)d_in[7];     // [2, E] disease->drug
    float* out = (float*)d_out_v;                    // [Nd+Ns, H]

    const int Nd = in_sizes[0] / H;   // 100000
    const int Ns = in_sizes[1] / H;   // 100000
    const int E  = in_sizes[6] / 2;   // 600000

    // workspace carve-out (256B aligned slices)
    char* wsp = (char*)d_ws;
    auto carve = [&](size_t bytes) {
        char* p = wsp;
        wsp += (bytes + 255) & ~(size_t)255;
        return p;
    };
    float* w0       = (float*)carve((size_t)H * H * sizeof(float));
    float* invdeg_s = (float*)carve((size_t)Ns * sizeof(float));
    float* invdeg_d = (float*)carve((size_t)Nd * sizeof(float));
    size_t nodeMax  = (size_t)(Nd > Ns ? Nd : Ns) * H * sizeof(float);
    float* y        = (float*)carve(nodeMax);                          // message buffer
    float* nxd      = (float*)carve((size_t)Nd * H * sizeof(float));   // layer-1 drug out
    float* nxs      = (float*)carve((size_t)Ns * H * sizeof(float));   // layer-1 disease out

    // degrees (same for both layers): deg over dst of each edge list
    hipMemsetAsync(invdeg_s, 0, (size_t)Ns * sizeof(float), stream);
    hipMemsetAsync(invdeg_d, 0, (size_t)Nd * sizeof(float), stream);
    deg_count_kernel<<<(E + 255) / 256, 256, 0, stream>>>(edd + E,  invdeg_s, E);
    deg_count_kernel<<<(E + 255) / 256, 256, 0, stream>>>(edsd + E, invdeg_d, E);
    deg_recip_kernel<<<(Ns + 255) / 256, 256, 0, stream>>>(invdeg_s, Ns);
    deg_recip_kernel<<<(Nd + 255) / 256, 256, 0, stream>>>(invdeg_d, Nd);

    const float* xd = emb_drug;
    const float* xs = emb_dis;
    const int tilesD = Nd / 16;   // 100000 = 6250 * 16 exactly
    const int tilesS = Ns / 16;
    const int scatterBlocks = (int)(((long)E * 32 + 255) / 256);

    for (int l = 0; l < 2; ++l) {
        w_combine_kernel<<<(H * H) / 256, 256, 0, stream>>>(
            basis + (size_t)l * 30 * H * H, comp + (size_t)l * 8 * 30, w0);

        float* od = (l == 0) ? nxd : out;                      // drug rows
        float* os = (l == 0) ? nxs : out + (size_t)Nd * H;     // disease rows
        const float* rt = root + (size_t)l * H * H;

        // self path: nx = x @ root
        gemm_wmma_kernel<<<(tilesD + 7) / 8, 256, 0, stream>>>(xd, rt, od, tilesD);
        gemm_wmma_kernel<<<(tilesS + 7) / 8, 256, 0, stream>>>(xs, rt, os, tilesS);

        // drug -> disease messages: y = x_drug @ w0, scatter-mean into os
        gemm_wmma_kernel<<<(tilesD + 7) / 8, 256, 0, stream>>>(xd, w0, y, tilesD);
        scatter_kernel<<<scatterBlocks, 256, 0, stream>>>(edd, edd + E, y, invdeg_s, os, E);

        // disease -> drug messages: y = x_dis @ w0, scatter-mean into od
        gemm_wmma_kernel<<<(tilesS + 7) / 8, 256, 0, stream>>>(xs, w0, y, tilesS);
        scatter_kernel<<<scatterBlocks, 256, 0, stream>>>(edsd, edsd + E, y, invdeg_d, od, E);

        // relu(nx + bias)
        finalize_kernel<<<(int)(((long)Nd * H + 255) / 256), 256, 0, stream>>>(
            od, bias + (size_t)l * H, (long)Nd * H);
        finalize_kernel<<<(int)(((long)Ns * H + 255) / 256), 256, 0, stream>>>(
            os, bias + (size_t)l * H, (long)Ns * H);

        xd = od;
        xs = os;
    }
}